// GNN_84421877170708
// MI455X (gfx1250) — compile-verified
//
#include <hip/hip_runtime.h>
#include <hip/hip_bf16.h>

typedef float v2f __attribute__((ext_vector_type(2)));
typedef float v4f __attribute__((ext_vector_type(4)));
typedef float v8f __attribute__((ext_vector_type(8)));

#define IN_NF   64
#define HIDN    128
#define EHW     64
#define KIN_NODE (HIDN + EHW)   // 192

__device__ __forceinline__ float silu_f(float x) {
    return x / (1.0f + __expf(-x));
}

__device__ __forceinline__ v8f splat8(float v) {
    v8f r = {v, v, v, v, v, v, v, v};
    return r;
}

// D = A(16x4, f32) * B(4x16, f32) + C  -- exact fp32 WMMA
__device__ __forceinline__ v8f wmma4(v2f a, v2f b, v8f c) {
    return __builtin_amdgcn_wmma_f32_16x16x4_f32(
        /*neg_a=*/false, a, /*neg_b=*/false, b,
        /*c_mod=*/(short)0, c, /*reuse_a=*/false, /*reuse_b=*/false);
}

// ---------------------------------------------------------------------------
// h_out[N,128] = h_in[N,64] @ W[64,128] + b      (embedding_in, linear only)
// ---------------------------------------------------------------------------
__global__ void embed_kernel(const float* __restrict__ hin,
                             const float* __restrict__ W,
                             const float* __restrict__ bias,
                             float* __restrict__ hout, int Nn) {
    const int lane = threadIdx.x & 31;
    const int c    = lane & 15;
    const int half = lane >> 4;
    const int nw   = (gridDim.x * blockDim.x) >> 5;
    const int wid0 = (blockIdx.x * blockDim.x + threadIdx.x) >> 5;
    const int tiles = (Nn + 15) >> 4;

    for (int t = wid0; t < tiles; t += nw) {
        int mr = t * 16 + c;
        if (mr >= Nn) mr = Nn - 1;

        v8f acc[8];
#pragma unroll
        for (int n = 0; n < 8; ++n) acc[n] = splat8(bias[n * 16 + c]);

        for (int k = 0; k < IN_NF; k += 4) {
            const int kk = k + half * 2;
            v2f a = *(const v2f*)(hin + (size_t)mr * IN_NF + kk);
#pragma unroll
            for (int n = 0; n < 8; ++n) {
                v2f b;
                b.x = W[kk * HIDN + n * 16 + c];
                b.y = W[(kk + 1) * HIDN + n * 16 + c];
                acc[n] = wmma4(a, b, acc[n]);
            }
        }
        if (t * 16 + 16 <= Nn) {                       // full tile (uniform)
#pragma unroll
            for (int n = 0; n < 8; ++n) {
                float* p = hout + (size_t)(t * 16 + half * 8) * HIDN + n * 16 + c;
#pragma unroll
                for (int i = 0; i < 8; ++i) p[(size_t)i * HIDN] = acc[n][i];
            }
        } else {
#pragma unroll
            for (int n = 0; n < 8; ++n)
#pragma unroll
                for (int i = 0; i < 8; ++i) {
                    const int r = t * 16 + i + half * 8;
                    if (r < Nn) hout[(size_t)r * HIDN + n * 16 + c] = acc[n][i];
                }
        }
    }
}

// ---------------------------------------------------------------------------
// Fused edge model + scatter. Weights staged in LDS once per block:
//   w1s stride 72 (2*72 = 16 mod 64 -> half-waves hit disjoint banks).
//   m = silu(silu(concat(h[row],h[col]) @ w1 + b1) @ w2 + b2); agg[row] += m
// ---------------------------------------------------------------------------
__global__ void edge_kernel(const float* __restrict__ h,
                            const int* __restrict__ erow,
                            const int* __restrict__ ecol,
                            const float* __restrict__ w1,
                            const float* __restrict__ b1,
                            const float* __restrict__ w2,
                            const float* __restrict__ b2,
                            float* __restrict__ agg, int Ee) {
    constexpr int WS  = 72;                       // weight LDS row stride
    constexpr int LWM = EHW + 4;                  // 68, m1 tile row stride
    __shared__ float w1s[2 * HIDN * WS];          // 73728 B
    __shared__ float w2s[EHW * WS];               // 18432 B
    __shared__ float m1s[8][16 * LWM];            // 34816 B  (total ~124 KB)

    // ---- cooperative weight staging (vectorized, row-pad preserved) ----
    for (int i4 = threadIdx.x; i4 < (2 * HIDN * EHW) / 4; i4 += blockDim.x) {
        const int r = i4 >> 4, col = (i4 & 15) * 4;
        *(v4f*)(w1s + r * WS + col) = *(const v4f*)(w1 + r * EHW + col);
    }
    for (int i4 = threadIdx.x; i4 < (EHW * EHW) / 4; i4 += blockDim.x) {
        const int r = i4 >> 4, col = (i4 & 15) * 4;
        *(v4f*)(w2s + r * WS + col) = *(const v4f*)(w2 + r * EHW + col);
    }
    __syncthreads();

    const int lane = threadIdx.x & 31;
    const int wv   = threadIdx.x >> 5;
    const int c    = lane & 15;
    const int half = lane >> 4;
    const int nw   = (gridDim.x * blockDim.x) >> 5;
    const int wid0 = (blockIdx.x * blockDim.x + threadIdx.x) >> 5;
    const int tiles = (Ee + 15) >> 4;
    float* L = m1s[wv];

    for (int t = wid0; t < tiles; t += nw) {
        int ea = t * 16 + c;
        if (ea >= Ee) ea = Ee - 1;
        const float* __restrict__ hra = h + (size_t)erow[ea] * HIDN;
        const float* __restrict__ hca = h + (size_t)ecol[ea] * HIDN;

        // ---- GEMM1: [16,256] x [256,64] ----
        v8f acc[4];
#pragma unroll
        for (int n = 0; n < 4; ++n) acc[n] = splat8(b1[n * 16 + c]);

#pragma unroll 4
        for (int k = 0; k < HIDN; k += 4) {           // first half: h[row]
            const int kk = k + half * 2;
            v2f a = *(const v2f*)(hra + kk);
#pragma unroll
            for (int n = 0; n < 4; ++n) {
                v2f b;
                b.x = w1s[kk * WS + n * 16 + c];
                b.y = w1s[(kk + 1) * WS + n * 16 + c];
                acc[n] = wmma4(a, b, acc[n]);
            }
        }
#pragma unroll 4
        for (int k = 0; k < HIDN; k += 4) {           // second half: h[col]
            const int kk = k + half * 2;
            v2f a = *(const v2f*)(hca + kk);
#pragma unroll
            for (int n = 0; n < 4; ++n) {
                v2f b;
                b.x = w1s[(HIDN + kk) * WS + n * 16 + c];
                b.y = w1s[(HIDN + kk + 1) * WS + n * 16 + c];
                acc[n] = wmma4(a, b, acc[n]);
            }
        }

        // ---- silu -> LDS (D-layout store, row-major tile) ----
#pragma unroll
        for (int n = 0; n < 4; ++n)
#pragma unroll
            for (int i = 0; i < 8; ++i)
                L[(i + half * 8) * LWM + n * 16 + c] = silu_f(acc[n][i]);

        asm volatile("s_wait_dscnt 0" ::: "memory");

        // ---- GEMM2: [16,64] x [64,64] ----
        v8f acc2[4];
#pragma unroll
        for (int n = 0; n < 4; ++n) acc2[n] = splat8(b2[n * 16 + c]);

#pragma unroll 4
        for (int k = 0; k < EHW; k += 4) {
            const int kk = k + half * 2;
            v2f a;
            a.x = L[c * LWM + kk];
            a.y = L[c * LWM + kk + 1];
#pragma unroll
            for (int n = 0; n < 4; ++n) {
                v2f b;
                b.x = w2s[kk * WS + n * 16 + c];
                b.y = w2s[(kk + 1) * WS + n * 16 + c];
                acc2[n] = wmma4(a, b, acc2[n]);
            }
        }

        // ---- silu + atomic scatter-add into agg[row] ----
        if (t * 16 + 16 <= Ee) {                      // full tile (uniform)
#pragma unroll
            for (int i = 0; i < 8; ++i) {
                float* dst = agg + (size_t)erow[t * 16 + i + half * 8] * EHW + c;
#pragma unroll
                for (int n = 0; n < 4; ++n)
                    atomicAdd(dst + n * 16, silu_f(acc2[n][i]));
            }
        } else {
#pragma unroll
            for (int i = 0; i < 8; ++i) {
                const int e = t * 16 + i + half * 8;
                if (e < Ee) {
                    float* dst = agg + (size_t)erow[e] * EHW + c;
#pragma unroll
                    for (int n = 0; n < 4; ++n)
                        atomicAdd(dst + n * 16, silu_f(acc2[n][i]));
                }
            }
        }
        asm volatile("s_wait_dscnt 0" ::: "memory");  // LDS WAR before next tile
    }
}

// ---------------------------------------------------------------------------
// Node model (in-place residual):
//   z = silu(concat(h,agg) @ m1 + c1) @ m2 + c2 ;  h += z
// nW2 staged in LDS (stride 136: 2*136 = 16 mod 64 -> disjoint bank halves),
// so GEMM2 is entirely LDS-fed.
// ---------------------------------------------------------------------------
__global__ void node_kernel(float* __restrict__ h,
                            const float* __restrict__ agg,
                            const float* __restrict__ m1,
                            const float* __restrict__ c1,
                            const float* __restrict__ m2,
                            const float* __restrict__ c2,
                            int Nn) {
    constexpr int LW  = HIDN + 4;                 // 132, z1 tile row stride
    constexpr int WS2 = HIDN + 8;                 // 136, nW2 LDS row stride
    __shared__ float w2s[HIDN * WS2];             // 69632 B
    __shared__ float lds[8][16 * LW];             // 67584 B (total ~134 KB)

    // ---- stage nW2[128,128] ----
    for (int i4 = threadIdx.x; i4 < (HIDN * HIDN) / 4; i4 += blockDim.x) {
        const int r = i4 >> 5, col = (i4 & 31) * 4;
        *(v4f*)(w2s + r * WS2 + col) = *(const v4f*)(m2 + r * HIDN + col);
    }
    __syncthreads();

    const int lane = threadIdx.x & 31;
    const int wv   = threadIdx.x >> 5;
    const int c    = lane & 15;
    const int half = lane >> 4;
    const int nw   = (gridDim.x * blockDim.x) >> 5;
    const int wid0 = (blockIdx.x * blockDim.x + threadIdx.x) >> 5;
    const int tiles = (Nn + 15) >> 4;
    float* L = lds[wv];

    for (int t = wid0; t < tiles; t += nw) {
        int mr = t * 16 + c;
        if (mr >= Nn) mr = Nn - 1;
        const float* __restrict__ hr = h + (size_t)mr * HIDN;
        const float* __restrict__ ar = agg + (size_t)mr * EHW;

        // ---- GEMM1: [16,192] x [192,128] ----
        v8f acc[8];
#pragma unroll
        for (int n = 0; n < 8; ++n) acc[n] = splat8(c1[n * 16 + c]);

        for (int k = 0; k < HIDN; k += 4) {           // h part
            const int kk = k + half * 2;
            v2f a = *(const v2f*)(hr + kk);
#pragma unroll
            for (int n = 0; n < 8; ++n) {
                v2f b;
                b.x = m1[kk * HIDN + n * 16 + c];
                b.y = m1[(kk + 1) * HIDN + n * 16 + c];
                acc[n] = wmma4(a, b, acc[n]);
            }
        }
        for (int k = 0; k < EHW; k += 4) {            // agg part
            const int kk = k + half * 2;
            v2f a = *(const v2f*)(ar + kk);
#pragma unroll
            for (int n = 0; n < 8; ++n) {
                v2f b;
                b.x = m1[(HIDN + kk) * HIDN + n * 16 + c];
                b.y = m1[(HIDN + kk + 1) * HIDN + n * 16 + c];
                acc[n] = wmma4(a, b, acc[n]);
            }
        }
#pragma unroll
        for (int n = 0; n < 8; ++n)
#pragma unroll
            for (int i = 0; i < 8; ++i)
                L[(i + half * 8) * LW + n * 16 + c] = silu_f(acc[n][i]);

        asm volatile("s_wait_dscnt 0" ::: "memory");

        // ---- GEMM2: [16,128] x [128,128] + residual (all-LDS operands) ----
        v8f acc2[8];
#pragma unroll
        for (int n = 0; n < 8; ++n) acc2[n] = splat8(c2[n * 16 + c]);

        for (int k = 0; k < HIDN; k += 4) {
            const int kk = k + half * 2;
            v2f a;
            a.x = L[c * LW + kk];
            a.y = L[c * LW + kk + 1];
#pragma unroll
            for (int n = 0; n < 8; ++n) {
                v2f b;
                b.x = w2s[kk * WS2 + n * 16 + c];
                b.y = w2s[(kk + 1) * WS2 + n * 16 + c];
                acc2[n] = wmma4(a, b, acc2[n]);
            }
        }
        if (t * 16 + 16 <= Nn) {                      // full tile (uniform)
#pragma unroll
            for (int n = 0; n < 8; ++n) {
                float* p = h + (size_t)(t * 16 + half * 8) * HIDN + n * 16 + c;
#pragma unroll
                for (int i = 0; i < 8; ++i) p[(size_t)i * HIDN] += acc2[n][i];
            }
        } else {
#pragma unroll
            for (int n = 0; n < 8; ++n)
#pragma unroll
                for (int i = 0; i < 8; ++i) {
                    const int r = t * 16 + i + half * 8;
                    if (r < Nn) h[(size_t)r * HIDN + n * 16 + c] += acc2[n][i];
                }
        }
        asm volatile("s_wait_dscnt 0" ::: "memory");
    }
}

// ---------------------------------------------------------------------------
// out[N,64] = h[N,128] @ W_out[128,64] + b_out
// ---------------------------------------------------------------------------
__global__ void out_kernel(const float* __restrict__ h,
                           const float* __restrict__ W,
                           const float* __restrict__ bias,
                           float* __restrict__ out, int Nn) {
    const int lane = threadIdx.x & 31;
    const int c    = lane & 15;
    const int half = lane >> 4;
    const int nw   = (gridDim.x * blockDim.x) >> 5;
    const int wid0 = (blockIdx.x * blockDim.x + threadIdx.x) >> 5;
    const int tiles = (Nn + 15) >> 4;

    for (int t = wid0; t < tiles; t += nw) {
        int mr = t * 16 + c;
        if (mr >= Nn) mr = Nn - 1;

        v8f acc[4];
#pragma unroll
        for (int n = 0; n < 4; ++n) acc[n] = splat8(bias[n * 16 + c]);

        for (int k = 0; k < HIDN; k += 4) {
            const int kk = k + half * 2;
            v2f a = *(const v2f*)(h + (size_t)mr * HIDN + kk);
#pragma unroll
            for (int n = 0; n < 4; ++n) {
                v2f b;
                b.x = W[kk * EHW + n * 16 + c];
                b.y = W[(kk + 1) * EHW + n * 16 + c];
                acc[n] = wmma4(a, b, acc[n]);
            }
        }
        if (t * 16 + 16 <= Nn) {                      // full tile (uniform)
#pragma unroll
            for (int n = 0; n < 4; ++n) {
                float* p = out + (size_t)(t * 16 + half * 8) * EHW + n * 16 + c;
#pragma unroll
                for (int i = 0; i < 8; ++i) p[(size_t)i * EHW] = acc[n][i];
            }
        } else {
#pragma unroll
            for (int n = 0; n < 4; ++n)
#pragma unroll
                for (int i = 0; i < 8; ++i) {
                    const int r = t * 16 + i + half * 8;
                    if (r < Nn) out[(size_t)r * EHW + n * 16 + c] = acc[n][i];
                }
        }
    }
}

static inline int imin_h(int a, int b) { return a < b ? a : b; }
static inline int cdiv_h(int a, int b) { return (a + b - 1) / b; }

extern "C" void kernel_launch(void* const* d_in, const int* in_sizes, int n_in,
                              void* d_out, int out_size, void* d_ws, size_t ws_size,
                              hipStream_t stream) {
    const float* h_in  = (const float*)d_in[0];
    const int*   edges = (const int*)d_in[1];
    const float* W_in  = (const float*)d_in[2];
    const float* b_in  = (const float*)d_in[3];
    const float* eW1   = (const float*)d_in[4];
    const float* eb1   = (const float*)d_in[5];
    const float* eW2   = (const float*)d_in[6];
    const float* eb2   = (const float*)d_in[7];
    const float* nW1   = (const float*)d_in[8];
    const float* nb1   = (const float*)d_in[9];
    const float* nW2   = (const float*)d_in[10];
    const float* nb2   = (const float*)d_in[11];
    const float* W_out = (const float*)d_in[12];
    const float* b_out = (const float*)d_in[13];

    const int Nn = in_sizes[0] / IN_NF;               // 50000
    const int Ee = in_sizes[1] / 2;                   // 800000
    const int Ll = in_sizes[4] / (2 * HIDN * EHW);    // 4

    float* h   = (float*)d_ws;                        // [N,128]
    float* agg = h + (size_t)Nn * HIDN;               // [N,64]

    const int* erow = edges;
    const int* ecol = edges + Ee;

    const int tilesN = cdiv_h(Nn, 16);
    const int tilesE = cdiv_h(Ee, 16);
    const int gEmb = imin_h(cdiv_h(tilesN, 8), 2048);
    const int gEdg = imin_h(cdiv_h(tilesE, 8), 2048);
    const int gNod = imin_h(cdiv_h(tilesN, 8), 2048);

    embed_kernel<<<gEmb, 256, 0, stream>>>(h_in, W_in, b_in, h, Nn);

    for (int l = 0; l < Ll; ++l) {
        hipMemsetAsync(agg, 0, (size_t)Nn * EHW * sizeof(float), stream);
        edge_kernel<<<gEdg, 256, 0, stream>>>(
            h, erow, ecol,
            eW1 + (size_t)l * 2 * HIDN * EHW, eb1 + (size_t)l * EHW,
            eW2 + (size_t)l * EHW * EHW,      eb2 + (size_t)l * EHW,
            agg, Ee);
        node_kernel<<<gNod, 256, 0, stream>>>(
            h, agg,
            nW1 + (size_t)l * KIN_NODE * HIDN, nb1 + (size_t)l * HIDN,
            nW2 + (size_t)l * HIDN * HIDN,     nb2 + (size_t)l * HIDN,
            Nn);
    }

    out_kernel<<<gEmb, 256, 0, stream>>>(h, W_out, b_out, (float*)d_out, Nn);
}